// DiffusionLoss_40226663694450
// MI455X (gfx1250) — compile-verified
//
#include <hip/hip_runtime.h>
#include <cfloat>

typedef __attribute__((ext_vector_type(2))) float v2f;
typedef __attribute__((ext_vector_type(8))) float v8f;

#define K_NB 9            // k+1 neighbors kept (includes self)
#define CSTAGE 128        // candidates staged in LDS per round
#define CONT_BLOCK 128    // 4 waves, 128 queries per block
#define TPAD 20           // transpose row pitch (floats): 16B-aligned rows, split bank groups

// ---------------- workspace layout (floats) ----------------
// ws[0]=recon sum  ws[1]=percep sum  ws[2]=cont sum  ws[3]=bnd sum  ws[4]=bnd count

__global__ void zero_ws_kernel(float* ws) {
  if (threadIdx.x < 8) ws[threadIdx.x] = 0.0f;
}

__device__ __forceinline__ float block_reduce_sum(float v) {
  __shared__ float red[32];
  const int lane = threadIdx.x & 31;
  const int wid  = threadIdx.x >> 5;
  #pragma unroll
  for (int o = 16; o > 0; o >>= 1) v += __shfl_down(v, o, 32);
  __syncthreads();                       // protect red[] reuse across calls
  if (lane == 0) red[wid] = v;
  __syncthreads();
  const int nw = (int)((blockDim.x + 31) >> 5);
  v = ((int)threadIdx.x < nw) ? red[threadIdx.x] : 0.0f;
  if (wid == 0) {
    #pragma unroll
    for (int o = 16; o > 0; o >>= 1) v += __shfl_down(v, o, 32);
  }
  return v;  // valid in thread 0
}

// ---------------- recon / percep: grid-stride MSE sum ----------------
__global__ void mse_sum_kernel(const float* __restrict__ a,
                               const float* __restrict__ b,
                               int n, float* __restrict__ acc) {
  float s = 0.0f;
  for (int i = blockIdx.x * blockDim.x + threadIdx.x; i < n;
       i += gridDim.x * blockDim.x) {
    float d = a[i] - b[i];
    s = fmaf(d, d, s);
  }
  s = block_reduce_sum(s);
  if (threadIdx.x == 0) atomicAdd(acc, s);
}

// Build per-wave WMMA fragments so D = d^2 directly:
//   A[M][:] = [-2x_M, -2y_M, -2z_M, 1]   (16x4 f32 A layout: V0=K0/K2, V1=K1/K3)
//   B[:][N] = [ x_N,   y_N,   z_N, |c_N|^2 ]
//   C[M][N] = |q_M|^2  (accumulator input, constant per wave)
//   => D[M][N] = |q_M|^2 + |c_N|^2 - 2 q_M . c_N
__device__ __forceinline__ void make_query_frags(
    float qx, float qy, float qz, float qn, int half, int l16,
    v2f& a0, v2f& a1, v8f& cq0, v8f& cq1) {
  const float x0 = __shfl(qx, l16, 32),      y0 = __shfl(qy, l16, 32),      z0 = __shfl(qz, l16, 32);
  const float x1 = __shfl(qx, 16 + l16, 32), y1 = __shfl(qy, 16 + l16, 32), z1 = __shfl(qz, 16 + l16, 32);
  a0.x = -2.0f * (half ? z0 : x0);  a0.y = half ? 1.0f : (-2.0f * y0);
  a1.x = -2.0f * (half ? z1 : x1);  a1.y = half ? 1.0f : (-2.0f * y1);
  #pragma unroll
  for (int r = 0; r < 8; ++r) {
    cq0[r] = __shfl(qn, r + half * 8, 32);        // C rows for queries 0..15
    cq1[r] = __shfl(qn, 16 + r + half * 8, 32);   // C rows for queries 16..31
  }
}

// ---------------- continuity: WMMA d^2 tiles + per-lane top-9 ----------------
__global__ __launch_bounds__(CONT_BLOCK)
void continuity_kernel(const float* __restrict__ pts, int B, int N,
                       float* __restrict__ ws) {
  const int blocksPerBatch = N / CONT_BLOCK;
  const int batch = blockIdx.x / blocksPerBatch;
  const int qbase = (blockIdx.x % blocksPerBatch) * CONT_BLOCK;
  const float* __restrict__ P = pts + (size_t)batch * (size_t)N * 3;

  __shared__ float4 stage[CSTAGE];                       // (x,y,z,|c|^2)
  __shared__ float  trans[CONT_BLOCK / 32][32 * TPAD];   // per-wave d^2 transpose

  const int lane = threadIdx.x & 31;
  const int wave = threadIdx.x >> 5;
  const int half = lane >> 4;
  const int l16  = lane & 15;

  const int q = qbase + wave * 32 + lane;
  const float qx = P[3 * q + 0];
  const float qy = P[3 * q + 1];
  const float qz = P[3 * q + 2];
  const float qn = fmaf(qx, qx, fmaf(qy, qy, qz * qz));

  v2f a0, a1; v8f cq0, cq1;
  make_query_frags(qx, qy, qz, qn, half, l16, a0, a1, cq0, cq1);

  float bd[K_NB]; int bi[K_NB];
  #pragma unroll
  for (int s = 0; s < K_NB; ++s) { bd[s] = FLT_MAX; bi[s] = 0; }

  auto insert9 = [&](float d2, int idx) {
    if (d2 < bd[K_NB - 1]) {
      bd[K_NB - 1] = d2; bi[K_NB - 1] = idx;
      #pragma unroll
      for (int s = K_NB - 1; s > 0; --s) {
        if (bd[s] < bd[s - 1]) {
          float td = bd[s]; bd[s] = bd[s - 1]; bd[s - 1] = td;
          int   ti = bi[s]; bi[s] = bi[s - 1]; bi[s - 1] = ti;
        }
      }
    }
  };

  float* __restrict__ tw = &trans[wave][0];

  for (int base = 0; base < N; base += CSTAGE) {
    __syncthreads();
    {
      const int j = base + threadIdx.x;          // CSTAGE == blockDim.x
      const float x = P[3 * j + 0];
      const float y = P[3 * j + 1];
      const float z = P[3 * j + 2];
      stage[threadIdx.x] = make_float4(x, y, z, fmaf(x, x, fmaf(y, y, z * z)));
      if (base + CSTAGE < N)                     // overlap next tile's HBM fetch
        __builtin_prefetch(&P[3 * (base + CSTAGE + threadIdx.x)], 0, 1);
    }
    __syncthreads();

    for (int sub = 0; sub < CSTAGE / 16; ++sub) {
      const float4 c = stage[sub * 16 + l16];
      v2f b;
      b.x = half ? c.z : c.x;
      b.y = half ? c.w : c.y;      // K=3 slot carries |c|^2

      v8f d0 = __builtin_amdgcn_wmma_f32_16x16x4_f32(false, a0, false, b,
                                                     (short)0, cq0, false, false);
      v8f d1 = __builtin_amdgcn_wmma_f32_16x16x4_f32(false, a1, false, b,
                                                     (short)0, cq1, false, false);

      // transpose d^2 tiles through wave-private LDS (DS ops in-order per wave)
      #pragma unroll
      for (int r = 0; r < 8; ++r) {
        tw[(r + half * 8) * TPAD + l16]      = d0[r];   // rows 0..15
        tw[(16 + r + half * 8) * TPAD + l16] = d1[r];   // rows 16..31
      }

      // each lane scans its own row: 4 x ds_load_b128, group-min gated insertion
      #pragma unroll
      for (int gg = 0; gg < 4; ++gg) {
        const float4 dv = *reinterpret_cast<const float4*>(&tw[lane * TPAD + 4 * gg]);
        const float gmin = fminf(fminf(dv.x, dv.y), fminf(dv.z, dv.w));
        if (gmin < bd[K_NB - 1]) {
          const int ib = base + sub * 16 + 4 * gg;
          insert9(dv.x, ib + 0);
          insert9(dv.y, ib + 1);
          insert9(dv.z, ib + 2);
          insert9(dv.w, ib + 3);
        }
      }
    }
  }

  // bd[0] is self (d~0). Neighbors = bi[1..8]: mean + sum of squared deviations.
  float nx[8], ny[8], nz[8];
  float mx = 0.f, my = 0.f, mz = 0.f;
  #pragma unroll
  for (int s = 0; s < 8; ++s) {
    const int j = bi[s + 1];
    nx[s] = P[3 * j + 0]; ny[s] = P[3 * j + 1]; nz[s] = P[3 * j + 2];
    mx += nx[s]; my += ny[s]; mz += nz[s];
  }
  mx *= 0.125f; my *= 0.125f; mz *= 0.125f;
  float loss = 0.0f;
  #pragma unroll
  for (int s = 0; s < 8; ++s) {
    const float dx = nx[s] - mx, dy = ny[s] - my, dz = nz[s] - mz;
    loss = fmaf(dx, dx, fmaf(dy, dy, fmaf(dz, dz, loss)));
  }
  const float tot = block_reduce_sum(loss);
  if (threadIdx.x == 0) atomicAdd(&ws[2], tot);
}

// ---------------- boundary: WMMA d^2 tiles + per-lane running min ----------------
__global__ __launch_bounds__(CONT_BLOCK)
void boundary_kernel(const float* __restrict__ c1, const float* __restrict__ c2,
                     int n1, int n2, float* __restrict__ ws) {
  __shared__ float4 stage[CSTAGE];
  __shared__ float  trans[CONT_BLOCK / 32][32 * TPAD];

  const int lane = threadIdx.x & 31;
  const int wave = threadIdx.x >> 5;
  const int half = lane >> 4;
  const int l16  = lane & 15;

  const int i = blockIdx.x * CONT_BLOCK + wave * 32 + lane;   // n1 % CONT_BLOCK == 0
  const float px = c1[3 * i + 0];
  const float py = c1[3 * i + 1];
  const float pz = c1[3 * i + 2];
  const float pn = fmaf(px, px, fmaf(py, py, pz * pz));

  v2f a0, a1; v8f cq0, cq1;
  make_query_frags(px, py, pz, pn, half, l16, a0, a1, cq0, cq1);

  float best = FLT_MAX;
  float* __restrict__ tw = &trans[wave][0];

  for (int base = 0; base < n2; base += CSTAGE) {
    __syncthreads();
    {
      const int j = base + threadIdx.x;
      const float x = c2[3 * j], y = c2[3 * j + 1], z = c2[3 * j + 2];
      stage[threadIdx.x] = make_float4(x, y, z, fmaf(x, x, fmaf(y, y, z * z)));
      if (base + CSTAGE < n2)
        __builtin_prefetch(&c2[3 * (base + CSTAGE + threadIdx.x)], 0, 1);
    }
    __syncthreads();

    for (int sub = 0; sub < CSTAGE / 16; ++sub) {
      const float4 c = stage[sub * 16 + l16];
      v2f b;
      b.x = half ? c.z : c.x;
      b.y = half ? c.w : c.y;

      v8f d0 = __builtin_amdgcn_wmma_f32_16x16x4_f32(false, a0, false, b,
                                                     (short)0, cq0, false, false);
      v8f d1 = __builtin_amdgcn_wmma_f32_16x16x4_f32(false, a1, false, b,
                                                     (short)0, cq1, false, false);
      #pragma unroll
      for (int r = 0; r < 8; ++r) {
        tw[(r + half * 8) * TPAD + l16]      = d0[r];
        tw[(16 + r + half * 8) * TPAD + l16] = d1[r];
      }
      #pragma unroll
      for (int gg = 0; gg < 4; ++gg) {
        const float4 dv = *reinterpret_cast<const float4*>(&tw[lane * TPAD + 4 * gg]);
        best = fminf(best, fminf(fminf(dv.x, dv.y), fminf(dv.z, dv.w)));
      }
    }
  }

  const float d    = sqrtf(fmaxf(best, 0.0f));
  const float in   = (d < 0.1f) ? 1.0f : 0.0f;
  const float ssum = block_reduce_sum(in * d);
  const float scnt = block_reduce_sum(in);
  if (threadIdx.x == 0) {
    atomicAdd(&ws[3], ssum);
    atomicAdd(&ws[4], scnt);
  }
}

// ---------------- finalize: 5 scalars ----------------
__global__ void finalize_kernel(const float* __restrict__ ws, float* __restrict__ out,
                                float invRecon, float invPercep, float invCont) {
  if (threadIdx.x == 0 && blockIdx.x == 0) {
    const float recon  = ws[0] * invRecon;
    const float percep = ws[1] * invPercep;
    const float cont   = ws[2] * invCont;
    const float cnt    = ws[4];
    const float bnd    = (cnt > 0.5f) ? (ws[3] / fmaxf(cnt, 1.0f)) : 0.0f;
    out[0] = recon;
    out[1] = percep;
    out[2] = cont;
    out[3] = bnd;
    out[4] = recon + 0.5f * percep + 0.5f * cont + 1.0f * bnd;
  }
}

extern "C" void kernel_launch(void* const* d_in, const int* in_sizes, int n_in,
                              void* d_out, int out_size, void* d_ws, size_t ws_size,
                              hipStream_t stream) {
  const float* predicted = (const float*)d_in[0];
  const float* target    = (const float*)d_in[1];
  const float* pfeat     = (const float*)d_in[2];
  const float* tfeat     = (const float*)d_in[3];
  const float* chunk1    = (const float*)d_in[4];
  const float* chunk2    = (const float*)d_in[5];
  float* ws  = (float*)d_ws;
  float* out = (float*)d_out;

  const int nRecon  = in_sizes[0];          // B*N*3 = 98304
  const int nPercep = in_sizes[2];          // B*F   = 4096
  const int B = 4;
  const int N = nRecon / (B * 3);           // 8192
  const int n1 = in_sizes[4] / 3;           // 4096
  const int n2 = in_sizes[5] / 3;           // 4096

  zero_ws_kernel<<<1, 32, 0, stream>>>(ws);

  mse_sum_kernel<<<128, 256, 0, stream>>>(predicted, target, nRecon, &ws[0]);
  mse_sum_kernel<<<8, 256, 0, stream>>>(pfeat, tfeat, nPercep, &ws[1]);

  continuity_kernel<<<B * (N / CONT_BLOCK), CONT_BLOCK, 0, stream>>>(predicted, B, N, ws);

  boundary_kernel<<<n1 / CONT_BLOCK, CONT_BLOCK, 0, stream>>>(chunk1, chunk2, n1, n2, ws);

  finalize_kernel<<<1, 32, 0, stream>>>(
      ws, out,
      1.0f / (float)nRecon,
      1.0f / (float)nPercep,
      1.0f / (float)(B * N * 8));
}